// LengthRegulator_24489903522412
// MI455X (gfx1250) — compile-verified
//
#include <hip/hip_runtime.h>
#include <hip/hip_bf16.h>
#include <stdint.h>

// Problem constants from the reference
#define B_ 32
#define S_ 1024
#define D_ 512

// Vector types matching the async-builtins' expected pointee types
typedef int v4i __attribute__((vector_size(16)));
typedef int v2i __attribute__((vector_size(8)));
typedef __attribute__((address_space(1))) v4i gv4i;   // global, 16B
typedef __attribute__((address_space(3))) v4i lv4i;   // LDS, 16B
typedef __attribute__((address_space(1))) v2i gv2i;   // global, 8B
typedef __attribute__((address_space(3))) v2i lv2i;   // LDS, 8B

#if __has_builtin(__builtin_amdgcn_global_load_async_to_lds_b128) && \
    __has_builtin(__builtin_amdgcn_global_store_async_from_lds_b64) && \
    __has_builtin(__builtin_amdgcn_s_wait_asynccnt)
#define USE_ASYNC_LDS 1
#else
#define USE_ASYNC_LDS 0
#endif

// ---------------------------------------------------------------------------
// Kernel 1: per-batch-row duration rounding + inclusive scan (S = 1024).
// One block per batch row, 1024 threads, Hillis-Steele scan in LDS.
// ---------------------------------------------------------------------------
__global__ __launch_bounds__(1024) void scan_kernel(
    const float* __restrict__ durations, const int* __restrict__ scale_p,
    int* __restrict__ ends, int* __restrict__ totals, float* __restrict__ tot_out) {
  __shared__ int sh[S_];
  const int b = blockIdx.x;
  const int t = threadIdx.x;
  const float scale = (float)(*scale_p);

  // d = clip(trunc(dur*scale + 0.5), 1, inf)   (durations >= 0, trunc == astype(int32))
  float v = durations[b * S_ + t] * scale + 0.5f;
  int d = (int)v;
  if (d < 1) d = 1;
  sh[t] = d;
  __syncthreads();

  for (int off = 1; off < S_; off <<= 1) {
    int add = (t >= off) ? sh[t - off] : 0;
    __syncthreads();
    sh[t] += add;
    __syncthreads();
  }

  ends[b * S_ + t] = sh[t];
  if (t == S_ - 1) {
    totals[b] = sh[t];
    tot_out[b] = (float)sh[t];  // second tuple output, stored as float
  }
}

// ---------------------------------------------------------------------------
// Kernel 2: expansion. One block per (b,s). Stage the 2KB source row into LDS
// once via the gfx1250 async Global->LDS DMA path, then fan out `d` copies
// with async LDS->Global stores (row read from global exactly once).
// ---------------------------------------------------------------------------
__global__ __launch_bounds__(256) void expand_kernel(
    const float* __restrict__ x, const int* __restrict__ ends,
    float* __restrict__ out, int T) {
  __shared__ float row[D_];  // 2048 bytes

  const int blk = blockIdx.x;
  const int b = blk >> 10;       // / S_
  const int s = blk & (S_ - 1);  // % S_
  const int tid = threadIdx.x;

  const int end_f   = ends[b * S_ + s];
  const int start_f = (s == 0) ? 0 : ends[b * S_ + s - 1];

  const float* src = x + ((size_t)(b * S_ + s)) * D_;
  float* dstbase   = out + (size_t)b * T * D_;

#if USE_ASYNC_LDS
  // 128 lanes x 16B = 2048B : async DMA global -> LDS, no VGPR roundtrip
  if (tid < 128) {
    gv4i* gsrc = (gv4i*)src + tid;
    lv4i* ldst = (lv4i*)&row[0] + tid;
    __builtin_amdgcn_global_load_async_to_lds_b128(gsrc, ldst, 0, 0);
  }
  __builtin_amdgcn_s_wait_asynccnt(0);
  __syncthreads();

  // 256 lanes x 8B = 2048B per replicated frame : async DMA LDS -> global
  lv2i* lsrc = (lv2i*)&row[0] + tid;
  for (int rep = start_f; rep < end_f; ++rep) {
    gv2i* gdst = (gv2i*)(dstbase + (size_t)rep * D_) + tid;
    __builtin_amdgcn_global_store_async_from_lds_b64(gdst, lsrc, 0, 0);
  }
  __builtin_amdgcn_s_wait_asynccnt(0);
#else
  // Fallback: register-staged vectorized copy (B128 load, B128 stores)
  float4 v4;
  if (tid < 128) v4 = ((const float4*)src)[tid];
  for (int rep = start_f; rep < end_f; ++rep) {
    if (tid < 128) ((float4*)(dstbase + (size_t)rep * D_))[tid] = v4;
  }
#endif
}

// ---------------------------------------------------------------------------
// Kernel 3: zero-fill padding frames [total_b, T). grid = (ceil(T/8), B).
// Each block covers 8 frames; early-exit if fully inside the valid region.
// ---------------------------------------------------------------------------
__global__ __launch_bounds__(256) void zero_pad_kernel(
    float* __restrict__ out, const int* __restrict__ totals, int T) {
  const int b = blockIdx.y;
  const int total = totals[b];
  const int f0 = blockIdx.x * 8;
  if (f0 + 8 <= total) return;

  const float2 z = make_float2(0.0f, 0.0f);
  for (int f = f0; f < f0 + 8 && f < T; ++f) {
    if (f >= total) {
      float2* p = (float2*)(out + ((size_t)b * T + f) * D_);
      p[threadIdx.x] = z;  // 256 lanes x 8B = one 2KB frame
    }
  }
}

// ---------------------------------------------------------------------------
extern "C" void kernel_launch(void* const* d_in, const int* in_sizes, int n_in,
                              void* d_out, int out_size, void* d_ws, size_t ws_size,
                              hipStream_t stream) {
  const float* x         = (const float*)d_in[0];
  const float* durations = (const float*)d_in[1];
  const int*   scale_p   = (const int*)d_in[2];
  float* out = (float*)d_out;

  // out = [B,T,D] frames + [B] totals, concatenated flat
  const int T = (out_size - B_) / (B_ * D_);

  int* ends   = (int*)d_ws;            // B_*S_ ints = 128 KB
  int* totals = ends + B_ * S_;        // B_ ints
  float* tot_out = out + (size_t)B_ * T * D_;

  scan_kernel<<<B_, S_, 0, stream>>>(durations, scale_p, ends, totals, tot_out);
  expand_kernel<<<B_ * S_, 256, 0, stream>>>(x, ends, out, T);
  zero_pad_kernel<<<dim3((T + 7) / 8, B_), 256, 0, stream>>>(out, totals, T);
}